// SO3GEUChebNet_86870008529173
// MI455X (gfx1250) — compile-verified
//
#include <hip/hip_runtime.h>

#define THREADS 256

static constexpr int BB = 4;                 // batch
static constexpr long long FFp[6]  = {12, 48, 192, 768, 3072, 12288};          // healpix pixels
static constexpr long long VVn[6]  = {72, 288, 1152, 4608, 18432, 73728};      // graph nodes (= F*6)
static constexpr long long EEn[6]  = {576, 2304, 9216, 36864, 147456, 589824}; // edges (= V*8)

typedef __attribute__((ext_vector_type(2))) float v2f;
typedef __attribute__((ext_vector_type(4))) float v4f;
typedef __attribute__((ext_vector_type(8))) float v8f;

#ifdef __has_builtin
#if __has_builtin(__builtin_amdgcn_wmma_f32_16x16x4_f32)
#define HAVE_WMMA_F32X4 1
#endif
#endif
#ifndef HAVE_WMMA_F32X4
#define HAVE_WMMA_F32X4 0
#endif

__device__ __forceinline__ void gAtomicAddF(float* p, float v) {
  (void)__hip_atomic_fetch_add(p, v, __ATOMIC_RELAXED, __HIP_MEMORY_SCOPE_AGENT);
}

// ---------------------------------------------------------------- elementwise
__global__ void k_fill_(float* __restrict__ p, long long n, float v) {
  long long t = (long long)blockIdx.x * blockDim.x + threadIdx.x;
  if (t < n) p[t] = v;
}

__global__ void k_scale_(const float* __restrict__ x, float* __restrict__ y,
                         long long n, float a) {
  long long t = (long long)blockIdx.x * blockDim.x + threadIdx.x;
  if (t < n) y[t] = a * x[t];
}

__global__ void k_relu_(float* __restrict__ x, long long n) {
  long long t = (long long)blockIdx.x * blockDim.x + threadIdx.x;
  if (t < n) x[t] = fmaxf(x[t], 0.f);
}

// (B,Csrc,V) channel-major -> node-major (v*B+b)*Cdst+c, zero-padding c>=Csrc
__global__ void k_tonm_(const float* __restrict__ x, float* __restrict__ out,
                        long long V, int Csrc, int Cdst) {
  long long total = V * BB * Cdst;
  long long t = (long long)blockIdx.x * blockDim.x + threadIdx.x;
  if (t >= total) return;
  int c = (int)(t % Cdst);
  long long r = t / Cdst;
  int b = (int)(r % BB);
  long long v = r / BB;
  out[t] = (c < Csrc) ? x[((long long)b * Csrc + c) * V + v] : 0.f;
}

// pad weights (K,Cs,Cos) -> (K,Cd,Cod) with zeros
__global__ void k_padw_(const float* __restrict__ src, float* __restrict__ dst,
                        int K, int Cs, int Cos, int Cd, int Cod) {
  long long total = (long long)K * Cd * Cod;
  long long t = (long long)blockIdx.x * blockDim.x + threadIdx.x;
  if (t >= total) return;
  int o = (int)(t % Cod);
  long long r = t / Cod;
  int cc = (int)(r % Cd);
  int k = (int)(r / Cd);
  dst[t] = (cc < Cs && o < Cos) ? src[((long long)k * Cs + cc) * Cos + o] : 0.f;
}

// ---------------------------------------------------------------- sparse Laplacian
// One thread per (edge, channel-quad): b128 gather + 4 coalesced f32 atomics.
// BC (=B*C) is always a multiple of 4; buffers are 16B aligned.
__global__ void k_spmv_(const int* __restrict__ rows, const int* __restrict__ cols,
                        const float* __restrict__ w, const float* __restrict__ x,
                        float* __restrict__ y, long long E, int BCq, int BC, float scale) {
  long long total = E * BCq;
  long long t = (long long)blockIdx.x * blockDim.x + threadIdx.x;
  if (t >= total) return;
  long long e = t / BCq;
  int r4 = (int)(t - e * BCq) * 4;
  float ws = scale * w[e];
  const v4f xv = *(const v4f*)(x + (long long)cols[e] * BC + r4);
  float* yp = y + (long long)rows[e] * BC + r4;
  gAtomicAddF(yp + 0, ws * xv.x);
  gAtomicAddF(yp + 1, ws * xv.y);
  gAtomicAddF(yp + 2, ws * xv.z);
  gAtomicAddF(yp + 3, ws * xv.w);
}

// ---------------------------------------------------------------- WMMA GEMM
// out[m*Cout+n] = (acc? out : bias/0) + sum_c A[m*Cin+c] * W[c*Cout+n]
// Requirements (guaranteed by padding): Cin%4==0, Cout%16==0, Mtot%16==0.
// Grid: blocks = mBlocks * ntN; all 8 waves of a block share one N tile, whose
// Cin x 16 weight slab is staged once in LDS (transposed, +2 pad -> conflict-free
// ds_load_b64 per lane). A operand is one aligned global b64 per lane per step.
// A (f32 16x4): lanes 0-15 K=0/1, lanes 16-31 K=2/3 (M = lane%16); B mirrored.
// C/D (f32 16x16): VGPR r -> M=r (lanes 0-15) / M=r+8 (lanes 16-31), N = lane%16.
__global__ __launch_bounds__(THREADS)
void k_gemm_(const float* __restrict__ A, int Cin, int Cinp,
             const float* __restrict__ W, const float* __restrict__ bias,
             float* __restrict__ out, int Cout, long long Mtot, int acc, int mBlocks) {
  extern __shared__ float lWt[];  // [16][Cinp] transposed weight tile
  const int ntM = (int)(Mtot >> 4);
  const int tn = (int)blockIdx.x / mBlocks;
  const int mc = (int)blockIdx.x % mBlocks;
  const int nb = tn << 4;

  for (int idx = (int)threadIdx.x; idx < Cin * 16; idx += THREADS) {
    int k = idx >> 4, nn = idx & 15;
    lWt[nn * Cinp + k] = W[(long long)k * Cout + nb + nn];
  }
  __syncthreads();

  const int tm = mc * (THREADS / 32) + ((int)threadIdx.x >> 5);
  if (tm >= ntM) return;
  const long long mb = (long long)tm << 4;
  const int lane = (int)threadIdx.x & 31;
  const int half = lane >> 4;
  const int l = lane & 15;
  const int n = nb + l;

  v8f c0 = {0.f, 0.f, 0.f, 0.f, 0.f, 0.f, 0.f, 0.f};
  v8f c1 = {0.f, 0.f, 0.f, 0.f, 0.f, 0.f, 0.f, 0.f};
  if (acc) {
#pragma unroll
    for (int r = 0; r < 8; ++r) c0[r] = out[(mb + r + half * 8) * (long long)Cout + n];
  } else if (bias != nullptr) {
    const float bv = bias[n];
#pragma unroll
    for (int r = 0; r < 8; ++r) c0[r] = bv;
  }

  const float* __restrict__ Arow = A + (mb + l) * (long long)Cin;
  const float* __restrict__ lrow = lWt + l * Cinp;

#if HAVE_WMMA_F32X4
  int kc = 0;
  for (; kc + 8 <= Cin; kc += 8) {
    const int k0 = kc + half * 2;
    const v2f a0 = *(const v2f*)(Arow + k0);
    const v2f b0 = *(const v2f*)(lrow + k0);
    c0 = __builtin_amdgcn_wmma_f32_16x16x4_f32(false, a0, false, b0, (short)0, c0, false, false);
    const int k1 = k0 + 4;
    const v2f a1 = *(const v2f*)(Arow + k1);
    const v2f b1 = *(const v2f*)(lrow + k1);
    c1 = __builtin_amdgcn_wmma_f32_16x16x4_f32(false, a1, false, b1, (short)0, c1, false, false);
  }
  if (kc < Cin) {  // Cin % 8 == 4 tail (e.g. Cin==4)
    const int k0 = kc + half * 2;
    const v2f a0 = *(const v2f*)(Arow + k0);
    const v2f b0 = *(const v2f*)(lrow + k0);
    c0 = __builtin_amdgcn_wmma_f32_16x16x4_f32(false, a0, false, b0, (short)0, c0, false, false);
  }
#else
  for (int k = 0; k < Cin; ++k) {
    const float wvv = lrow[k];
#pragma unroll
    for (int r = 0; r < 8; ++r)
      c0[r] += A[(mb + r + half * 8) * (long long)Cin + k] * wvv;
  }
#endif

  const v8f cf = c0 + c1;
#pragma unroll
  for (int r = 0; r < 8; ++r) out[(mb + r + half * 8) * (long long)Cout + n] = cf[r];
}

// ---------------------------------------------------------------- batchnorm
__global__ void k_bnstats_(const float* __restrict__ x, long long total, int C,
                           float* __restrict__ sum, float* __restrict__ sumsq) {
  __shared__ float s1[512];
  __shared__ float s2[512];
  for (int i = threadIdx.x; i < C; i += blockDim.x) { s1[i] = 0.f; s2[i] = 0.f; }
  __syncthreads();
  for (long long t = (long long)blockIdx.x * blockDim.x + threadIdx.x; t < total;
       t += (long long)gridDim.x * blockDim.x) {
    float v = x[t];
    int c = (int)(t % C);
    atomicAdd(&s1[c], v);
    atomicAdd(&s2[c], v * v);
  }
  __syncthreads();
  for (int i = threadIdx.x; i < C; i += blockDim.x) {
    if (s1[i] != 0.f) gAtomicAddF(&sum[i], s1[i]);
    if (s2[i] != 0.f) gAtomicAddF(&sumsq[i], s2[i]);
  }
}

__global__ void k_bnapply_(float* __restrict__ x, const float* __restrict__ sum,
                           const float* __restrict__ sumsq, long long Nn, int C,
                           const float* __restrict__ gm, const float* __restrict__ bt,
                           const float* __restrict__ res, int relu) {
  long long total = Nn * (long long)C;
  long long t = (long long)blockIdx.x * blockDim.x + threadIdx.x;
  if (t >= total) return;
  int c = (int)(t % C);
  float inv = 1.f / (float)Nn;
  float m = sum[c] * inv;
  float var = fmaxf(sumsq[c] * inv - m * m, 0.f);
  float v = (x[t] - m) * rsqrtf(var + 1e-5f) * gm[c] + bt[c];
  if (res) v += res[t];
  if (relu) v = fmaxf(v, 0.f);
  x[t] = v;
}

// ---------------------------------------------------------------- pool / unpool+concat / ori-pool
__global__ void k_pool_(const float* __restrict__ in, float* __restrict__ out,
                        long long Fc, int C) {  // Fc = coarse pixel count
  long long total = Fc * 6 * BB * C;
  long long t = (long long)blockIdx.x * blockDim.x + threadIdx.x;
  if (t >= total) return;
  int c = (int)(t % C);
  long long r = t / C;
  int b = (int)(r % BB);
  long long v = r / BB;  // coarse node = pix*6+ori
  long long pix = v / 6;
  int ori = (int)(v % 6);
  float m = -3.402823466e38f;
#pragma unroll
  for (int j = 0; j < 4; ++j) {
    long long vin = (pix * 4 + j) * 6 + ori;
    m = fmaxf(m, in[(vin * BB + b) * (long long)C + c]);
  }
  out[t] = m;
}

// out = concat(unpool(up from coarse), skip) at fine level; C = C1+C2
__global__ void k_catup_(const float* __restrict__ up, const float* __restrict__ skip,
                         float* __restrict__ out, long long Vfine, int C1, int C2) {
  int C = C1 + C2;
  long long total = Vfine * BB * C;
  long long t = (long long)blockIdx.x * blockDim.x + threadIdx.x;
  if (t >= total) return;
  int c = (int)(t % C);
  long long r = t / C;
  int b = (int)(r % BB);
  long long v = r / BB;  // fine node = pix*6+ori
  float val;
  if (c < C1) {
    long long pix = v / 6;
    int ori = (int)(v % 6);
    long long vc = (pix >> 2) * 6 + ori;  // coarse parent node
    val = up[(vc * BB + b) * (long long)C1 + c];
  } else {
    val = skip[(v * BB + b) * (long long)C2 + (c - C1)];
  }
  out[t] = val;
}

__global__ void k_oripool_(const float* __restrict__ in, float* __restrict__ out,
                           long long F, int C) {
  long long total = F * BB * C;
  long long t = (long long)blockIdx.x * blockDim.x + threadIdx.x;
  if (t >= total) return;
  int c = (int)(t % C);
  long long r = t / C;
  int b = (int)(r % BB);
  long long f = r / BB;
  float m = -3.402823466e38f;
#pragma unroll
  for (int o = 0; o < 6; ++o)
    m = fmaxf(m, in[((f * 6 + o) * BB + b) * (long long)C + c]);
  out[t] = m;
}

// logits node-major (f*B+b, LD) -> log_softmax over first CO, write (B, CO, F)
__global__ void k_lsm_(const float* __restrict__ logits, float* __restrict__ out,
                       long long F, int LD) {
  const int CO = 10;
  long long t = (long long)blockIdx.x * blockDim.x + threadIdx.x;
  if (t >= F * BB) return;
  int b = (int)(t % BB);
  long long f = t / BB;
  const float* p = logits + t * LD;
  float mx = p[0];
#pragma unroll
  for (int o = 1; o < CO; ++o) mx = fmaxf(mx, p[o]);
  float s = 0.f;
#pragma unroll
  for (int o = 0; o < CO; ++o) s += __expf(p[o] - mx);
  float lse = mx + __logf(s);
#pragma unroll
  for (int o = 0; o < CO; ++o)
    out[((long long)b * CO + o) * F + f] = p[o] - lse;
}

// ================================================================ host side
namespace {

struct Arena {
  char* base;
  size_t cap;
  size_t off;
  float* alloc(long long nflo) {
    size_t bytes = ((size_t)nflo * sizeof(float) + 255) & ~(size_t)255;
    float* p = (float*)(base + off);
    off += bytes;
    return p;
  }
};

struct Graph {
  const int* rows;
  const int* cols;
  const float* w;
  long long E;
};

struct Blk {
  const float *w1, *b1, *w2, *b2, *g1, *be1, *g2, *be2, *wres;
};

inline unsigned gfor(long long total) {
  long long b = (total + THREADS - 1) / THREADS;
  return b < 1 ? 1u : (unsigned)b;
}

Blk getblk(void* const* din, int& i, bool wres) {
  Blk b;
  b.w1 = (const float*)din[i++]; b.b1 = (const float*)din[i++];
  b.w2 = (const float*)din[i++]; b.b2 = (const float*)din[i++];
  b.g1 = (const float*)din[i++]; b.be1 = (const float*)din[i++];
  b.g2 = (const float*)din[i++]; b.be2 = (const float*)din[i++];
  b.wres = wres ? (const float*)din[i++] : nullptr;
  return b;
}

void gemm(hipStream_t s, const float* A, int Cin, const float* W, const float* bias,
          float* out, int Cout, long long Mtot, int acc) {
  const int ntM = (int)(Mtot >> 4);
  const int ntN = Cout >> 4;                    // Cout is a multiple of 16
  const int mBlocks = (ntM + (THREADS / 32) - 1) / (THREADS / 32);
  const int Cinp = Cin + 2;                     // LDS bank-conflict padding
  const size_t shmem = (size_t)16 * Cinp * sizeof(float);
  k_gemm_<<<(unsigned)(mBlocks * ntN), THREADS, shmem, s>>>(A, Cin, Cinp, W, bias, out,
                                                            Cout, Mtot, acc, mBlocks);
}

void spmv(hipStream_t s, const Graph& g, const float* x, float* y, int C, float scale) {
  const int BC = BB * C;            // always a multiple of 4
  const int BCq = BC >> 2;
  long long total = g.E * (long long)BCq;
  k_spmv_<<<gfor(total), THREADS, 0, s>>>(g.rows, g.cols, g.w, x, y, g.E, BCq, BC, scale);
}

// out = sum_k T_k @ W[k] (+bias);  T0=x, T1=Lx, Tk=2*L*T(k-1)-T(k-2)
void cheb(hipStream_t s, const Graph& g, const float* x, int Cin, const float* W,
          const float* bias, float* out, int Cout, long long Nn, Arena& ar) {
  size_t mark = ar.off;
  long long sz = Nn * (long long)Cin;
  float* Ta = ar.alloc(sz);
  float* Tb = ar.alloc(sz);
  long long WS = (long long)Cin * Cout;
  gemm(s, x, Cin, W, bias, out, Cout, Nn, 0);                       // k=0
  k_fill_<<<gfor(sz), THREADS, 0, s>>>(Ta, sz, 0.f);
  spmv(s, g, x, Ta, Cin, 1.f);                                      // T1 = L x
  gemm(s, Ta, Cin, W + WS, nullptr, out, Cout, Nn, 1);
  k_scale_<<<gfor(sz), THREADS, 0, s>>>(x, Tb, sz, -1.f);           // Tb = -T0
  spmv(s, g, Ta, Tb, Cin, 2.f);                                     // T2 = 2L T1 - T0
  gemm(s, Tb, Cin, W + 2 * WS, nullptr, out, Cout, Nn, 1);
  k_scale_<<<gfor(sz), THREADS, 0, s>>>(Ta, Ta, sz, -1.f);          // Ta = -T1
  spmv(s, g, Tb, Ta, Cin, 2.f);                                     // T3 = 2L T2 - T1
  gemm(s, Ta, Cin, W + 3 * WS, nullptr, out, Cout, Nn, 1);
  ar.off = mark;
}

void bn(hipStream_t s, float* xbuf, long long Nn, int C, const float* gm,
        const float* bt, const float* res, int relu, float* stats) {
  k_fill_<<<gfor(1024), THREADS, 0, s>>>(stats, 1024, 0.f);
  long long total = Nn * (long long)C;
  unsigned blocks = gfor(total);
  if (blocks > 4096u) blocks = 4096u;
  k_bnstats_<<<blocks, THREADS, 0, s>>>(xbuf, total, C, stats, stats + 512);
  k_bnapply_<<<gfor(total), THREADS, 0, s>>>(xbuf, stats, stats + 512, Nn, C, gm, bt, res, relu);
}

void res_block(hipStream_t s, const Graph& g, const Blk& p, const float* xin,
               int Cin, int Cout, long long Nn, float* out, Arena& ar, float* stats) {
  size_t mark = ar.off;
  float* h1 = ar.alloc(Nn * (long long)Cout);
  cheb(s, g, xin, Cin, p.w1, p.b1, h1, Cout, Nn, ar);
  bn(s, h1, Nn, Cout, p.g1, p.be1, nullptr, 1, stats);
  cheb(s, g, h1, Cout, p.w2, p.b2, out, Cout, Nn, ar);
  const float* sc = xin;
  if (p.wres) {
    float* scb = ar.alloc(Nn * (long long)Cout);
    gemm(s, xin, Cin, p.wres, nullptr, scb, Cout, Nn, 0);
    sc = scb;
  }
  bn(s, out, Nn, Cout, p.g2, p.be2, sc, 1, stats);
  ar.off = mark;
}

}  // namespace

extern "C" void kernel_launch(void* const* d_in, const int* in_sizes, int n_in,
                              void* d_out, int out_size, void* d_ws, size_t ws_size,
                              hipStream_t stream) {
  (void)in_sizes; (void)n_in; (void)out_size;
  int i = 0;
  const float* x_in = (const float*)d_in[i++];
  const float* ciw = (const float*)d_in[i++];
  const float* cib = (const float*)d_in[i++];
  Blk enc5 = getblk(d_in, i, true);
  Blk enc4 = getblk(d_in, i, true);
  Blk enc3 = getblk(d_in, i, true);
  Blk enc2 = getblk(d_in, i, true);
  Blk enc1 = getblk(d_in, i, false);
  Blk enc0 = getblk(d_in, i, false);
  Blk dec1 = getblk(d_in, i, true);
  Blk dec2 = getblk(d_in, i, true);
  Blk dec3 = getblk(d_in, i, true);
  Blk dec4 = getblk(d_in, i, true);
  Blk dec5 = getblk(d_in, i, true);
  const float* cow = (const float*)d_in[i++];
  Graph G[6];
  for (int g = 0; g < 6; ++g) {
    G[g].rows = (const int*)d_in[i++];
    G[g].cols = (const int*)d_in[i++];
    G[g].w = (const float*)d_in[i++];
    G[g].E = EEn[g];
  }

  long long NN[6];
  for (int l = 0; l < 6; ++l) NN[l] = VVn[l] * BB;

  Arena ar{(char*)d_ws, ws_size, 0};
  float* stats = ar.alloc(1024);

  hipStream_t s = stream;

  // padded weights: conv_in (4,3,16)->(4,4,16); head (1,16,10)->(1,16,16)
  float* ciwp = ar.alloc(4 * 4 * 16);
  k_padw_<<<gfor(4 * 4 * 16), THREADS, 0, s>>>(ciw, ciwp, 4, 3, 16, 4, 16);
  float* cowp = ar.alloc(16 * 16);
  k_padw_<<<gfor(16 * 16), THREADS, 0, s>>>(cow, cowp, 1, 16, 10, 16, 16);

  // input -> node-major, padded to 4 channels (4th channel is zero everywhere
  // and stays zero through the Laplacian recursion)
  float* xnm = ar.alloc(NN[5] * 4);
  k_tonm_<<<gfor(NN[5] * 4), THREADS, 0, s>>>(x_in, xnm, VVn[5], 3, 4);

  // conv_in + relu
  float* h = ar.alloc(NN[5] * 16);
  cheb(s, G[5], xnm, 4, ciwp, cib, h, 16, NN[5], ar);
  k_relu_<<<gfor(NN[5] * 16), THREADS, 0, s>>>(h, NN[5] * 16);

  // -------- encoder --------
  float* e5 = ar.alloc(NN[5] * 32);
  res_block(s, G[5], enc5, h, 16, 32, NN[5], e5, ar, stats);
  float* p5 = ar.alloc(NN[4] * 32);
  k_pool_<<<gfor(NN[4] * 32), THREADS, 0, s>>>(e5, p5, FFp[4], 32);

  float* e4 = ar.alloc(NN[4] * 64);
  res_block(s, G[4], enc4, p5, 32, 64, NN[4], e4, ar, stats);
  float* p4 = ar.alloc(NN[3] * 64);
  k_pool_<<<gfor(NN[3] * 64), THREADS, 0, s>>>(e4, p4, FFp[3], 64);

  float* e3 = ar.alloc(NN[3] * 128);
  res_block(s, G[3], enc3, p4, 64, 128, NN[3], e3, ar, stats);
  float* p3 = ar.alloc(NN[2] * 128);
  k_pool_<<<gfor(NN[2] * 128), THREADS, 0, s>>>(e3, p3, FFp[2], 128);

  float* e2 = ar.alloc(NN[2] * 256);
  res_block(s, G[2], enc2, p3, 128, 256, NN[2], e2, ar, stats);
  float* p2 = ar.alloc(NN[1] * 256);
  k_pool_<<<gfor(NN[1] * 256), THREADS, 0, s>>>(e2, p2, FFp[1], 256);

  float* e1 = ar.alloc(NN[1] * 256);
  res_block(s, G[1], enc1, p2, 256, 256, NN[1], e1, ar, stats);
  float* p1 = ar.alloc(NN[0] * 256);
  k_pool_<<<gfor(NN[0] * 256), THREADS, 0, s>>>(e1, p1, FFp[0], 256);

  float* e0 = ar.alloc(NN[0] * 256);
  res_block(s, G[0], enc0, p1, 256, 256, NN[0], e0, ar, stats);

  // -------- decoder with skip concatenation --------
  float* u1 = ar.alloc(NN[1] * 512);
  k_catup_<<<gfor(NN[1] * 512), THREADS, 0, s>>>(e0, e1, u1, VVn[1], 256, 256);
  float* d1 = ar.alloc(NN[1] * 256);
  res_block(s, G[1], dec1, u1, 512, 256, NN[1], d1, ar, stats);

  float* u2 = ar.alloc(NN[2] * 512);
  k_catup_<<<gfor(NN[2] * 512), THREADS, 0, s>>>(d1, e2, u2, VVn[2], 256, 256);
  float* d2 = ar.alloc(NN[2] * 128);
  res_block(s, G[2], dec2, u2, 512, 128, NN[2], d2, ar, stats);

  float* u3 = ar.alloc(NN[3] * 256);
  k_catup_<<<gfor(NN[3] * 256), THREADS, 0, s>>>(d2, e3, u3, VVn[3], 128, 128);
  float* d3 = ar.alloc(NN[3] * 64);
  res_block(s, G[3], dec3, u3, 256, 64, NN[3], d3, ar, stats);

  float* u4 = ar.alloc(NN[4] * 128);
  k_catup_<<<gfor(NN[4] * 128), THREADS, 0, s>>>(d3, e4, u4, VVn[4], 64, 64);
  float* d4 = ar.alloc(NN[4] * 32);
  res_block(s, G[4], dec4, u4, 128, 32, NN[4], d4, ar, stats);

  float* u5 = ar.alloc(NN[5] * 64);
  k_catup_<<<gfor(NN[5] * 64), THREADS, 0, s>>>(d4, e5, u5, VVn[5], 32, 32);
  float* d5 = ar.alloc(NN[5] * 16);
  res_block(s, G[5], dec5, u5, 64, 16, NN[5], d5, ar, stats);

  // -------- head: orientation max-pool, 16->10 (padded to 16) conv, log_softmax --------
  long long Mhead = FFp[5] * BB;  // 49152 (f,b) rows
  float* pori = ar.alloc(Mhead * 16);
  k_oripool_<<<gfor(Mhead * 16), THREADS, 0, s>>>(d5, pori, FFp[5], 16);
  float* logits = ar.alloc(Mhead * 16);
  gemm(s, pori, 16, cowp, nullptr, logits, 16, Mhead, 0);
  k_lsm_<<<gfor(Mhead), THREADS, 0, s>>>(logits, (float*)d_out, FFp[5], 16);
}